// Kernel_7886596238811945167_53472342835833
// MI455X (gfx1250) — compile-verified
//
#include <hip/hip_runtime.h>

typedef __attribute__((ext_vector_type(2))) float v2f;
typedef __attribute__((ext_vector_type(8))) float v8f;

#define NB   16
#define CC   256
#define WW   56
#define WP   60       // padded width (2 zeros each side)
#define HWp  3136     // 56*56
#define XPC  3360     // 56*60, padded plane per channel
#define KK   768      // 3*C
#define HALF_PI 1.5707963f

// ---------------- Kernel P: xp[n,c,h,wp] = (2<=wp<58) ? x[n,c,h,wp-2] : 0 -------------
__global__ __launch_bounds__(256) void k_pad(const float* __restrict__ x,
                                             float* __restrict__ xp) {
    const int t = blockIdx.x * 256 + threadIdx.x;     // over NB*CC*56*60
    const int wp   = t % WP;
    const int row  = t / WP;                          // n*CC*56 + c*56 + h
    const int w    = wp - 2;
    float v = 0.0f;
    if (w >= 0 && w < WW) v = x[(size_t)row * WW + w];
    xp[t] = v;
}

// ---------------- Kernel A: t3[n,o,p] = sum_k W3[o,k] * t1[n,k,p] ----------------
// GEMM M=256(o) N=3136(p) K=768. K blocked by 24 (2 c4-steps = 6 WMMAs), dual accumulators,
// hoisted tap base pointers -> immediate-offset loads in the steady state.
__global__ __launch_bounds__(128) void k_t3(const float* __restrict__ xp,
                                            const float* __restrict__ W3,
                                            float* __restrict__ t3) {
    const int lane = threadIdx.x & 31;
    const int wave = threadIdx.x >> 5;
    const int l16  = lane & 15;
    const int half = lane >> 4;
    const int p0   = (blockIdx.x * 4 + wave) * 16;
    const int o0   = blockIdx.y * 16;
    const int n    = blockIdx.z;

    int offx[3], offy[3];
#pragma unroll
    for (int j = 0; j < 3; ++j) {
        const int m0 = 4 * j + 2 * half, m1 = m0 + 1;
        offx[j] = (m0 / 3) * XPC + 2 * (m0 % 3);
        offy[j] = (m1 / 3) * XPC + 2 * (m1 % 3);
    }

    const int pcol = p0 + l16;
    const int pb60 = (pcol / WW) * WP + (pcol % WW);  // h*60 + w (one-time div)
    const float* xpb   = xp + (size_t)n * CC * XPC + pb60;
    const float* w3row = W3 + (size_t)(o0 + l16) * KK + 2 * half;

    const float* BX[3] = { xpb + offx[0], xpb + offx[1], xpb + offx[2] };
    const float* BY[3] = { xpb + offy[0], xpb + offy[1], xpb + offy[2] };

    v8f acc0 = {0.f,0.f,0.f,0.f,0.f,0.f,0.f,0.f};
    v8f acc1 = acc0;
    for (int c4 = 0; c4 < 64; c4 += 2) {
        const int kb = c4 * 12;
        const int xb = c4 * 4 * XPC;
#pragma unroll
        for (int u = 0; u < 2; ++u) {
#pragma unroll
            for (int j = 0; j < 3; ++j) {
                v2f a = *(const v2f*)(w3row + kb + 12 * u + 4 * j);
                v2f b;
                b.x = BX[j][xb + u * 4 * XPC];
                b.y = BY[j][xb + u * 4 * XPC];
                if (((u * 3 + j) & 1) == 0)
                    acc0 = __builtin_amdgcn_wmma_f32_16x16x4_f32(false, a, false, b, (short)0, acc0, false, false);
                else
                    acc1 = __builtin_amdgcn_wmma_f32_16x16x4_f32(false, a, false, b, (short)0, acc1, false, false);
            }
        }
    }
    acc0 = acc0 + acc1;

    float* t3_n = t3 + (size_t)n * CC * HWp;
    const int obase = o0 + half * 8;
#pragma unroll
    for (int r = 0; r < 8; ++r)
        t3_n[(obase + r) * HWp + pcol] = acc0[r];
}

// ---------------- Kernel B: t7[n,k,co] = p7w[k,co] * (1/56) * sum_p t2[n,k,p]*x[n,co,p] ----
// GEMM M=768(k) N=256(co) K=3136(p). Wave = 64(k) x 16(co); 4 independent accumulators,
// fixed rolled-channel row pointers, all fragments are contiguous b64 loads.
__global__ __launch_bounds__(128) void k_t7(const float* __restrict__ x,
                                            const float* __restrict__ xp,
                                            const float* __restrict__ p7w,
                                            float* __restrict__ t7) {
    const int lane = threadIdx.x & 31;
    const int wave = threadIdx.x >> 5;
    const int l16  = lane & 15;
    const int half = lane >> 4;
    const int id   = blockIdx.x * 4 + wave;     // 0..191
    const int k0   = (id % 12) * 64;
    const int co0  = (id / 12) * 16;
    const int n    = blockIdx.y;
    const float* x_n  = x  + (size_t)n * CC * HWp;
    const float* xp_n = xp + (size_t)n * CC * XPC;

    const float* arow[4];
#pragma unroll
    for (int i = 0; i < 4; ++i) {
        const int k   = k0 + 16 * i + l16;
        const int c2r = ((k / 3) + CC - 1) & (CC - 1);   // channel roll
        arow[i] = xp_n + (size_t)c2r * XPC + 2 * (k % 3) + 2 * half;
    }
    const float* brow = x_n + (size_t)(co0 + l16) * HWp + 2 * half;

    v8f acc[4];
#pragma unroll
    for (int i = 0; i < 4; ++i) acc[i] = (v8f){0.f,0.f,0.f,0.f,0.f,0.f,0.f,0.f};

    for (int h = 0; h < WW; ++h) {
        const int b56 = h * WW;
        const int b60 = h * WP;
#pragma unroll 2
        for (int w0 = 0; w0 < WW; w0 += 4) {     // K step of 4 never crosses a row
            v2f b = *(const v2f*)(brow + b56 + w0);
#pragma unroll
            for (int i = 0; i < 4; ++i) {
                v2f a = *(const v2f*)(arow[i] + b60 + w0);
                acc[i] = __builtin_amdgcn_wmma_f32_16x16x4_f32(false, a, false, b, (short)0, acc[i], false, false);
            }
        }
    }

    float* t7_n = t7 + (size_t)n * KK * CC;
    const float s = 1.0f / 56.0f;               // 1/sqrt(hw)
    const int co = co0 + l16;
#pragma unroll
    for (int i = 0; i < 4; ++i) {
        const int kbase = k0 + 16 * i + half * 8;
#pragma unroll
        for (int r = 0; r < 8; ++r) {
            const int k = kbase + r;
            t7_n[k * CC + co] = acc[i][r] * s * p7w[k * CC + co];
        }
    }
}

// ---------------- Kernel C: out[n,co,p] = (1/sqrt(768)) * sum_k t6T[p,k]*t7[n,k,co] --------
// t6 = max(t3, sin(pi/2 * t1)) on the fly; hoisted bases, dual accumulators.
__global__ __launch_bounds__(128) void k_out(const float* __restrict__ xp,
                                             const float* __restrict__ t3,
                                             const float* __restrict__ t7,
                                             float* __restrict__ out) {
    const int lane = threadIdx.x & 31;
    const int wave = threadIdx.x >> 5;
    const int l16  = lane & 15;
    const int half = lane >> 4;
    const int p0   = (blockIdx.x * 4 + wave) * 16;
    const int co0  = blockIdx.y * 16;
    const int n    = blockIdx.z;

    int offx[3], offy[3], t3ox[3], t3oy[3];
#pragma unroll
    for (int j = 0; j < 3; ++j) {
        const int m0 = 4 * j + 2 * half, m1 = m0 + 1;
        offx[j] = (m0 / 3) * XPC + 2 * (m0 % 3);
        offy[j] = (m1 / 3) * XPC + 2 * (m1 % 3);
        t3ox[j] = (m0 / 3) * HWp;
        t3oy[j] = (m1 / 3) * HWp;
    }

    const int prow = p0 + l16;
    const int pb60 = (prow / WW) * WP + (prow % WW);
    const float* xpb = xp + (size_t)n * CC * XPC + pb60;
    const float* t3b = t3 + (size_t)n * CC * HWp + prow;
    const float* t7b = t7 + (size_t)n * KK * CC + co0 + l16 + 2 * half * CC;

    const float* SX[3] = { xpb + offx[0], xpb + offx[1], xpb + offx[2] };
    const float* SY[3] = { xpb + offy[0], xpb + offy[1], xpb + offy[2] };
    const float* TX[3] = { t3b + t3ox[0], t3b + t3ox[1], t3b + t3ox[2] };
    const float* TY[3] = { t3b + t3oy[0], t3b + t3oy[1], t3b + t3oy[2] };

    v8f acc0 = {0.f,0.f,0.f,0.f,0.f,0.f,0.f,0.f};
    v8f acc1 = acc0;
    for (int c4 = 0; c4 < 64; c4 += 2) {
        const int kb = c4 * 12;
        const int xb = c4 * 4 * XPC;
        const int tb = c4 * 4 * HWp;
#pragma unroll
        for (int u = 0; u < 2; ++u) {
#pragma unroll
            for (int j = 0; j < 3; ++j) {
                const int xo = xb + u * 4 * XPC;
                const int to = tb + u * 4 * HWp;
                const int ko = kb + 12 * u + 4 * j;
                v2f a, b;
                a.x = fmaxf(TX[j][to], __sinf(HALF_PI * SX[j][xo]));
                a.y = fmaxf(TY[j][to], __sinf(HALF_PI * SY[j][xo]));
                b.x = t7b[ko * CC];
                b.y = t7b[(ko + 1) * CC];
                if (((u * 3 + j) & 1) == 0)
                    acc0 = __builtin_amdgcn_wmma_f32_16x16x4_f32(false, a, false, b, (short)0, acc0, false, false);
                else
                    acc1 = __builtin_amdgcn_wmma_f32_16x16x4_f32(false, a, false, b, (short)0, acc1, false, false);
            }
        }
    }
    acc0 = acc0 + acc1;

    float* out_n = out + (size_t)n * CC * HWp;
    const float scale = 0.03608439182435161f;   // 1/sqrt(768)
    const int co    = co0 + l16;
    const int pbase = p0 + half * 8;
#pragma unroll
    for (int r = 0; r < 8; ++r)                  // 8 consecutive p per lane
        out_n[co * HWp + pbase + r] = acc0[r] * scale;
}

extern "C" void kernel_launch(void* const* d_in, const int* in_sizes, int n_in,
                              void* d_out, int out_size, void* d_ws, size_t ws_size,
                              hipStream_t stream) {
    const float* x   = (const float*)d_in[0];   // (16,256,56,56)
    const float* W3  = (const float*)d_in[1];   // (256,768)
    const float* p7w = (const float*)d_in[2];   // (1,256,3,256) -> [k,co]
    float* out = (float*)d_out;

    float* xp = (float*)d_ws;                               // 16*256*56*60 f32 (55.0 MB)
    float* t3 = xp + (size_t)NB * CC * XPC;                 // 16*256*3136 f32 (51.4 MB)
    float* t7 = t3 + (size_t)NB * CC * HWp;                 // 16*768*256  f32 (12.6 MB)

    // P: width-pad x (zero borders) -> branch-free unfold taps downstream
    k_pad<<<dim3((NB * CC * WW * WP) / 256), 256, 0, stream>>>(x, xp);
    // A: t3   grid = (196/4 p-tiles, 16 o-tiles, 16 n), 4 waves/block
    k_t3 <<<dim3(49, 16, 16), 128, 0, stream>>>(xp, W3, t3);
    // B: t7   192 wave-tiles per n (12 k-tiles * 16 co-tiles)
    k_t7 <<<dim3(48, 16), 128, 0, stream>>>(x, xp, p7w, t7);
    // C: out  grid = (196/4 p-tiles, 16 co-tiles, 16 n)
    k_out<<<dim3(49, 16, 16), 128, 0, stream>>>(xp, t3, t7, out);
}